// ADM_18391049961898
// MI455X (gfx1250) — compile-verified
//
#include <hip/hip_runtime.h>

typedef __attribute__((ext_vector_type(2))) float v2f;
typedef __attribute__((ext_vector_type(8))) float v8f;

#define LSTRIDE 49            // 48 data cols + 1 pad (bank-conflict avoidance)
#define PLANE   518400LL      // 540*960 max per-batch plane stride for all scale buffers

// db2 taps as literal constants (kept out of memory -> pure cndmask selection)
#define LO0 ( 0.48296291314469f)
#define LO1 ( 0.836516303737469f)
#define LO2 ( 0.224143868041857f)
#define LO3 (-0.129409522550921f)
#define HI0 (-0.129409522550921f)
#define HI1 (-0.224143868041857f)
#define HI2 ( 0.836516303737469f)
#define HI3 (-0.48296291314469f)

__constant__ float cRF[4][3] = {
  {0.017382f, 0.017382f, 0.005891f},
  {0.031985f, 0.031985f, 0.014299f},
  {0.043373f, 0.043373f, 0.024397f},
  {0.045673f, 0.045673f, 0.031313f}};

__device__ __forceinline__ int refl(int i, int n) {
  if (i < 0) i = -i;
  if (i >= n) i = 2*n - 2 - i;
  if (i < 0) i = 0;
  if (i > n - 1) i = n - 1;
  return i;
}

// Branchless band-matrix tap: F[m][k] = f[k-2m] if 0 <= k-2m < 4 else 0.
// Lowered to v_cndmask chains over inline float literals — no memory access.
__device__ __forceinline__ float tapSel(int t, float f0, float f1, float f2, float f3) {
  float v = 0.0f;
  v = (t == 0) ? f0 : v;
  v = (t == 1) ? f1 : v;
  v = (t == 2) ? f2 : v;
  v = (t == 3) ? f3 : v;
  return v;
}
__device__ __forceinline__ float bandLO(int k, int m) { return tapSel(k - 2*m, LO0, LO1, LO2, LO3); }
__device__ __forceinline__ float bandHI(int k, int m) { return tapSel(k - 2*m, HI0, HI1, HI2, HI3); }

// ---------------------------------------------------------------------------
// Fused 2-D db2 DWT (stride-2 separable, reflect padding) for one 16x16 output
// tile per wave, both passes on the fp32 WMMA pipe (V_WMMA_F32_16X16X4_F32).
//   vertical:   out(16x16) = F(16x36 band) x X(36x16)   (3 column chunks)
//   horizontal: out(16x16) = Y(16x36)      x G(36x16 band), 4 subbands
// ---------------------------------------------------------------------------
__global__ void __launch_bounds__(32) dwt2_wmma(
    const float* __restrict__ src,
    float* __restrict__ outA, float* __restrict__ outH,
    float* __restrict__ outV, float* __restrict__ outD,
    int Hin, int Win, int Hout, int Wout,
    long long inPlane, int tilesX)
{
  __shared__ float ldsIn[36 * LSTRIDE];   // input patch (rows 2*ty*16-1 .., cols 2*tx*16-1 ..)
  __shared__ float ldsLo[16 * LSTRIDE];   // vertical lo_v result, 16 x 48
  __shared__ float ldsHi[16 * LSTRIDE];   // vertical hi_v result, 16 x 48

  const int lane   = threadIdx.x;
  const int tile   = blockIdx.x;
  const int b      = blockIdx.y;
  const int tx     = tile % tilesX, ty = tile / tilesX;
  const int N      = lane & 15;           // M (A rows) / N (B,C,D cols) per ISA layout
  const int hiHalf = lane >> 4;           // K half-select

  const float* in = src + (long long)b * inPlane;
  const int baseR = 32*ty - 1, baseC = 32*tx - 1;

  __builtin_prefetch(in + (long long)refl(baseR, Hin) * Win + refl(baseC, Win), 0, 1);

  // Stage 36x48 input patch into LDS with reflect-padding folded into indexing.
  for (int idx = lane; idx < 36*48; idx += 32) {
    int rr = idx / 48, cc = idx - rr*48;
    int gr = refl(baseR + rr, Hin);
    int gc = refl(baseC + cc, Win);
    ldsIn[rr*LSTRIDE + cc] = in[(long long)gr * Win + gc];
  }

  // ---- Vertical pass ----
  #pragma unroll
  for (int fsel = 0; fsel < 2; ++fsel) {           // static after unroll
    float* dst = fsel ? ldsHi : ldsLo;
    #pragma unroll
    for (int c = 0; c < 3; ++c) {
      v8f acc = {};
      #pragma unroll
      for (int kb = 0; kb < 9; ++kb) {
        const int k0 = 4*kb + 2*hiHalf;
        v2f A, Bm;
        A.x  = fsel ? bandHI(k0,     N) : bandLO(k0,     N);   // filter band matrix
        A.y  = fsel ? bandHI(k0 + 1, N) : bandLO(k0 + 1, N);
        Bm.x = ldsIn[k0      *LSTRIDE + c*16 + N];             // input columns
        Bm.y = ldsIn[(k0 + 1)*LSTRIDE + c*16 + N];
        acc = __builtin_amdgcn_wmma_f32_16x16x4_f32(false, A, false, Bm,
                                                    (short)0, acc, false, false);
      }
      #pragma unroll
      for (int r = 0; r < 8; ++r)
        dst[(r + 8*hiHalf)*LSTRIDE + c*16 + N] = acc[r];       // C/D layout -> LDS
    }
  }

  // ---- Horizontal pass ----
  #pragma unroll
  for (int o = 0; o < 4; ++o) {                    // (lo,LO)=a (hi,LO)=h (lo,HI)=v (hi,HI)=d
    const float* Y   = (o == 0 || o == 2) ? (const float*)ldsLo : (const float*)ldsHi;
    const bool   useHI = (o >= 2);
    v8f acc = {};
    #pragma unroll
    for (int kb = 0; kb < 9; ++kb) {
      const int k0 = 4*kb + 2*hiHalf;
      v2f A, Bm;
      A.x  = Y[N*LSTRIDE + k0];                                // vertical-pass rows
      A.y  = Y[N*LSTRIDE + k0 + 1];
      Bm.x = useHI ? bandHI(k0,     N) : bandLO(k0,     N);    // transposed filter band
      Bm.y = useHI ? bandHI(k0 + 1, N) : bandLO(k0 + 1, N);
      acc = __builtin_amdgcn_wmma_f32_16x16x4_f32(false, A, false, Bm,
                                                  (short)0, acc, false, false);
    }
    float* out = (o == 0 ? outA : (o == 1 ? outH : (o == 2 ? outV : outD)))
                 + (long long)b * PLANE;
    #pragma unroll
    for (int r = 0; r < 8; ++r) {
      int gy = ty*16 + r + 8*hiHalf;
      int gx = tx*16 + N;
      if (gy < Hout && gx < Wout)
        out[(long long)gy * Wout + gx] = acc[r];
    }
  }
}

// ---------------------------------------------------------------------------
// Decouple (restoration / additive split) per reference semantics.
// ---------------------------------------------------------------------------
__device__ __forceinline__ void decouple(
    float oh, float ov, float od, float th, float tv, float td,
    float& Rh, float& Rv, float& Rd, float& Ah, float& Av, float& Ad)
{
  const float EPS = 1e-30f, GAIN = 100.0f, C1 = 0.99969541350954785f; // cos(pi/180)^2
  float Kh = fminf(fmaxf(th / (oh + EPS), 0.f), 1.f);
  float Kv = fminf(fmaxf(tv / (ov + EPS), 0.f), 1.f);
  float Kd = fminf(fmaxf(td / (od + EPS), 0.f), 1.f);
  Rh = Kh*oh; Rv = Kv*ov; Rd = Kd*od;
  float ot = oh*th + ov*tv;
  bool flag = (ot >= 0.f) && (ot*ot >= C1*(oh*oh + ov*ov)*(th*th + tv*tv));
  if (flag) {
    if (Rh > 0.f) Rh = fminf(Rh*GAIN, th);
    if (Rh < 0.f) Rh = fmaxf(Rh*GAIN, th);
    if (Rv > 0.f) Rv = fminf(Rv*GAIN, tv);
    if (Rv < 0.f) Rv = fmaxf(Rv*GAIN, tv);
    if (Rd > 0.f) Rd = fminf(Rd*GAIN, td);
    if (Rd < 0.f) Rd = fmaxf(Rd*GAIN, td);
  }
  Ah = th - Rh; Av = tv - Rv; Ad = td - Rd;
}

// ---------------------------------------------------------------------------
// Fused decouple + contrast-masking (3x3) + border mask + |x|^3 pooling.
// Per-subband partial sums accumulated with global fp32 atomics.
// ---------------------------------------------------------------------------
__global__ void __launch_bounds__(256) adm_cm_reduce(
    const float* __restrict__ pOh, const float* __restrict__ pOv, const float* __restrict__ pOd,
    const float* __restrict__ pTh, const float* __restrict__ pTv, const float* __restrict__ pTd,
    float* __restrict__ accum, int Hs, int Ws, int scale, int tilesX)
{
  __shared__ float sA[3][18*19];     // |r*A| with 1-px reflect halo
  __shared__ float red[256*6];

  const int tid  = threadIdx.x;
  const int tile = blockIdx.x;
  const int b    = blockIdx.y;
  const int tx   = tile % tilesX, ty = tile / tilesX;
  const long long off = (long long)b * PLANE;
  const float r0 = cRF[scale][0], r1 = cRF[scale][1], r2 = cRF[scale][2];

  for (int i = tid; i < 18*18; i += 256) {
    int ly = i / 18, lx = i - ly*18;
    int gy = refl(ty*16 + ly - 1, Hs);
    int gx = refl(tx*16 + lx - 1, Ws);
    long long p = off + (long long)gy * Ws + gx;
    float Rh,Rv,Rd,Ah,Av,Ad;
    decouple(pOh[p],pOv[p],pOd[p],pTh[p],pTv[p],pTd[p],Rh,Rv,Rd,Ah,Av,Ad);
    sA[0][ly*19+lx] = fabsf(r0*Ah);
    sA[1][ly*19+lx] = fabsf(r1*Av);
    sA[2][ly*19+lx] = fabsf(r2*Ad);
  }
  __syncthreads();

  float vals[6] = {0.f,0.f,0.f,0.f,0.f,0.f};
  const int lx = tid & 15, ly = tid >> 4;
  const int gy = ty*16 + ly, gx = tx*16 + lx;
  if (gy < Hs && gx < Ws) {
    const int left = (int)floorf(0.1f*Ws - 0.5f);
    const int top  = (int)floorf(0.1f*Hs - 0.5f);
    bool mask = (gx < left) || (gx >= Ws - left) || (gy < top) || (gy >= Hs - top);
    if (!mask) {
      long long p = off + (long long)gy * Ws + gx;
      float Rh,Rv,Rd,Ah,Av,Ad;
      decouple(pOh[p],pOv[p],pOd[p],pTh[p],pTv[p],pTd[p],Rh,Rv,Rd,Ah,Av,Ad);
      float Rb[3] = {r0*Rh, r1*Rv, r2*Rd};
      float Ob[3] = {r0*pOh[p], r1*pOv[p], r2*pOd[p]};
      #pragma unroll
      for (int j = 0; j < 3; ++j) {
        float s9 = 0.f;
        #pragma unroll
        for (int dy = 0; dy < 3; ++dy)
          #pragma unroll
          for (int dx = 0; dx < 3; ++dx)
            s9 += sA[j][(ly+dy)*19 + (lx+dx)];
        // CM kernel = (ones3x3 + center)/30
        float thr = (s9 + sA[j][(ly+1)*19 + (lx+1)]) * (1.0f/30.0f);
        float n  = fmaxf(fabsf(Rb[j]) - thr, 0.f);
        float dd = fabsf(Ob[j]);
        vals[j*2+0] = n*n*n;
        vals[j*2+1] = dd*dd*dd;
      }
    }
  }

  #pragma unroll
  for (int j = 0; j < 6; ++j) red[tid*6+j] = vals[j];
  __syncthreads();
  for (int s = 128; s > 0; s >>= 1) {
    if (tid < s)
      #pragma unroll
      for (int j = 0; j < 6; ++j) red[tid*6+j] += red[(tid+s)*6+j];
    __syncthreads();
  }
  if (tid == 0) {
    float* dst = accum + (long long)(b*4 + scale) * 6;
    #pragma unroll
    for (int j = 0; j < 6; ++j) atomicAdd(&dst[j], red[j]);
  }
}

__global__ void zero_acc(float* a, int n) {
  int i = blockIdx.x*blockDim.x + threadIdx.x;
  if (i < n) a[i] = 0.f;
}

__global__ void adm_final(const float* __restrict__ accum, float* __restrict__ out,
                          int h0,int h1,int h2,int h3,int w0,int w1,int w2,int w3)
{
  int b = threadIdx.x;
  if (b >= 4) return;
  int hh[4] = {h0,h1,h2,h3}, ww[4] = {w0,w1,w2,w3};
  float num = 0.f, den = 0.f;
  for (int s = 0; s < 4; ++s) {
    int left = (int)floorf(0.1f*ww[s] - 0.5f);
    int top  = (int)floorf(0.1f*hh[s] - 0.5f);
    float extra = 3.0f * cbrtf((float)((hh[s]-2*top)*(ww[s]-2*left)) / 32.0f);
    const float* a = accum + (b*4 + s)*6;
    for (int j = 0; j < 3; ++j) { num += cbrtf(a[j*2+0]); den += cbrtf(a[j*2+1]); }
    num += extra; den += extra;
  }
  const float lim = 1e-10f;   // 1e-10 * (W*H)/(1920*1080) with W,H = 1920,1080
  if (num < lim) num = 0.f;
  if (den < lim) den = 0.f;
  out[b] = (den == 0.f) ? 1.f : num / den;
}

extern "C" void kernel_launch(void* const* d_in, const int* in_sizes, int n_in,
                              void* d_out, int out_size, void* d_ws, size_t ws_size,
                              hipStream_t stream)
{
  (void)n_in; (void)out_size; (void)ws_size;
  const int H0 = 1080, W0 = 1920;
  const int B  = in_sizes[0] / (H0 * W0);      // 4

  float* base = (float*)d_ws;
  const long long BP = (long long)B * PLANE;
  float* appR[2] = {base + 0*BP, base + 1*BP};
  float* appD[2] = {base + 2*BP, base + 3*BP};
  float* Oh = base + 4*BP; float* Ov = base + 5*BP; float* Od = base + 6*BP;
  float* Th = base + 7*BP; float* Tv = base + 8*BP; float* Td = base + 9*BP;
  float* accum = base + 10*BP;
  const int ACCN = B * 4 * 6;

  zero_acc<<<1, 128, 0, stream>>>(accum, ACCN);

  // dwt2 output dims per scale (reflect pad top=1, bottom=pb; left=1, right=pr)
  int hsArr[4], wsArr[4];
  {
    int hi = H0, wi = W0;
    for (int s = 0; s < 4; ++s) {
      int pb = (hi % 2 == 0) ? 1 : 2;
      int pr = (wi % 2 == 0) ? 1 : 2;
      hsArr[s] = (hi + pb - 3)/2 + 1;   // 540,270,135,68
      wsArr[s] = (wi + pr - 3)/2 + 1;   // 960,480,240,120
      hi = hsArr[s]; wi = wsArr[s];
    }
  }

  const float* curR = (const float*)d_in[0];
  const float* curD = (const float*)d_in[1];
  long long inPlane = (long long)H0 * W0;
  int hi = H0, wi = W0, pp = 0;

  for (int s = 0; s < 4; ++s) {
    int ho = hsArr[s], wo = wsArr[s];
    int tX = (wo + 15)/16, tY = (ho + 15)/16;
    dim3 grid(tX * tY, B);
    dwt2_wmma<<<grid, 32, 0, stream>>>(curR, appR[pp], Oh, Ov, Od, hi, wi, ho, wo, inPlane, tX);
    dwt2_wmma<<<grid, 32, 0, stream>>>(curD, appD[pp], Th, Tv, Td, hi, wi, ho, wo, inPlane, tX);
    adm_cm_reduce<<<grid, 256, 0, stream>>>(Oh, Ov, Od, Th, Tv, Td, accum, ho, wo, s, tX);
    curR = appR[pp]; curD = appD[pp]; pp ^= 1;
    hi = ho; wi = wo; inPlane = PLANE;
  }

  adm_final<<<1, 4, 0, stream>>>(accum, (float*)d_out,
                                 hsArr[0], hsArr[1], hsArr[2], hsArr[3],
                                 wsArr[0], wsArr[1], wsArr[2], wsArr[3]);
}